// RNN_16020228014402
// MI455X (gfx1250) — compile-verified
//
#include <hip/hip_runtime.h>
#include <hip/hip_bf16.h>
#include <math.h>

typedef float v2f __attribute__((ext_vector_type(2)));
typedef float v4f __attribute__((ext_vector_type(4)));
typedef float v8f __attribute__((ext_vector_type(8)));

#define HID 512
#define G3H 1536  // 3*HID

// ---------------------------------------------------------------------------
// Kernel 1: gi = w_ih @ x + b_ih ; gh = w_hh @ h + b_hh  via V_WMMA_F32_16X16X4_F32
//
// Latency-bound regime (6.3 MB total traffic << 23 MB bandwidth-delay product
// at 23.3 TB/s), so maximize concurrent loads: one 256-thread block (8 waves)
// per 16-row output tile, each wave owning a K=64 chunk -> 1536 waves put the
// entire weight stream in flight in one latency shadow.
//
// Grid: 192 blocks. Blocks 0..95 -> ih (vec = emb row), 96..191 -> hh (hidden).
// A tile layout (32-bit A 16x4): lane holds W[m_base + (lane&15)][k .. k+1]
// with k sub-offset koff = 2*(lane>=16). B replicates vec across all 16
// columns -> same per-lane k indexing. C/D layout: VGPR r holds row
// m_base + r (+8 for lanes 16..31). Per-wave partials are reduced across the
// 8 K-chunks in LDS (deterministic, no atomics); bias folded at final store.
// ---------------------------------------------------------------------------
__global__ __launch_bounds__(256) void gru_gates_wmma(
    const int*   __restrict__ token,
    const float* __restrict__ hidden,
    const float* __restrict__ emb,
    const float* __restrict__ w_ih,
    const float* __restrict__ w_hh,
    const float* __restrict__ b_ih,
    const float* __restrict__ b_hh,
    float*       __restrict__ gi,
    float*       __restrict__ gh)
{
    __shared__ float vec_s[HID];
    __shared__ float part[8][16];   // [k-chunk wave][row in tile]

    const int  tid   = threadIdx.x;
    const int  blk   = blockIdx.x;        // 0..191
    const bool is_hh = (blk >= 96);

    const float* W   = is_hh ? w_hh : w_ih;
    const float* bia = is_hh ? b_hh : b_ih;
    float*       out = is_hh ? gh   : gi;
    const float* vec = is_hh ? hidden : (emb + (size_t)token[0] * HID);

    // Stage the 512-float input vector into LDS (256 threads x float2).
    ((v2f*)vec_s)[tid] = ((const v2f*)vec)[tid];
    __syncthreads();

    const int wave   = tid >> 5;                      // 0..7 = K-chunk id (wave32)
    const int lane   = tid & 31;
    const int tile   = is_hh ? blk - 96 : blk;        // 0..95
    const int m_base = tile * 16;                     // output row base
    const int mrow   = lane & 15;                     // A-matrix row within tile
    const int koff   = (lane >> 4) << 1;              // 0 or 2 (A/B k sub-offset)
    const int crow   = (lane >> 4) << 3;              // C/D row base within tile

    const float* wrow = W + (size_t)(m_base + mrow) * HID + wave * 64 + koff;
    const float* vsub = vec_s + wave * 64 + koff;

    v8f acc0 = {}, acc1 = {}, acc2 = {}, acc3 = {};

    // K-chunk loop: 64 / 16 = 4 iterations, 4 independent WMMAs per iteration.
    #pragma unroll
    for (int k = 0; k < 64; k += 16) {
        v2f a0 = *(const v2f*)(wrow + k);
        v2f b0 = *(const v2f*)(vsub + k);
        acc0 = __builtin_amdgcn_wmma_f32_16x16x4_f32(false, a0, false, b0,
                                                     (short)0, acc0, false, false);
        v2f a1 = *(const v2f*)(wrow + k + 4);
        v2f b1 = *(const v2f*)(vsub + k + 4);
        acc1 = __builtin_amdgcn_wmma_f32_16x16x4_f32(false, a1, false, b1,
                                                     (short)0, acc1, false, false);
        v2f a2 = *(const v2f*)(wrow + k + 8);
        v2f b2 = *(const v2f*)(vsub + k + 8);
        acc2 = __builtin_amdgcn_wmma_f32_16x16x4_f32(false, a2, false, b2,
                                                     (short)0, acc2, false, false);
        v2f a3 = *(const v2f*)(wrow + k + 12);
        v2f b3 = *(const v2f*)(vsub + k + 12);
        acc3 = __builtin_amdgcn_wmma_f32_16x16x4_f32(false, a3, false, b3,
                                                     (short)0, acc3, false, false);
    }

    v8f acc = (acc0 + acc1) + (acc2 + acc3);

    // All 16 N-columns hold identical dot products; column 0 of each half
    // publishes this wave's K-chunk partial into LDS.
    if ((lane & 15) == 0) {
        #pragma unroll
        for (int r = 0; r < 8; ++r)
            part[wave][crow + r] = acc[r];
    }
    __syncthreads();

    // Threads 0..15: reduce the 8 K-chunk partials per row, add bias, store.
    if (tid < 16) {
        float sum = bia[m_base + tid];
        #pragma unroll
        for (int w = 0; w < 8; ++w)
            sum += part[w][tid];
        out[m_base + tid] = sum;
    }
}

// ---------------------------------------------------------------------------
// Kernel 2: gate nonlinearities, h_new, FC(2x512) via LDS reduction, log-softmax.
// One block of 512 threads.
// ---------------------------------------------------------------------------
__global__ __launch_bounds__(512) void gru_tail(
    const float* __restrict__ hidden,
    const float* __restrict__ fc_w,
    const float* __restrict__ fc_b,
    const float* __restrict__ gi,
    const float* __restrict__ gh,
    float*       __restrict__ out)
{
    __shared__ float s0[HID];
    __shared__ float s1[HID];

    const int i = threadIdx.x;

    const float r = 1.0f / (1.0f + expf(-(gi[i]       + gh[i])));
    const float z = 1.0f / (1.0f + expf(-(gi[HID + i] + gh[HID + i])));
    const float n = tanhf(gi[2 * HID + i] + r * gh[2 * HID + i]);
    const float hn = (1.0f - z) * n + z * hidden[i];

    s0[i] = hn * fc_w[i];
    s1[i] = hn * fc_w[HID + i];
    __syncthreads();

    for (int s = HID / 2; s > 0; s >>= 1) {
        if (i < s) {
            s0[i] += s0[i + s];
            s1[i] += s1[i + s];
        }
        __syncthreads();
    }

    if (i == 0) {
        const float l0 = s0[0] + fc_b[0];
        const float l1 = s1[0] + fc_b[1];
        const float m  = fmaxf(l0, l1);
        const float lse = m + logf(expf(l0 - m) + expf(l1 - m));
        out[0] = l0 - lse;
        out[1] = l1 - lse;
    }
}

// ---------------------------------------------------------------------------
extern "C" void kernel_launch(void* const* d_in, const int* in_sizes, int n_in,
                              void* d_out, int out_size, void* d_ws, size_t ws_size,
                              hipStream_t stream) {
    const int*   token  = (const int*)  d_in[0];
    const float* hidden = (const float*)d_in[1];
    const float* emb    = (const float*)d_in[2];
    const float* w_ih   = (const float*)d_in[3];
    const float* w_hh   = (const float*)d_in[4];
    const float* b_ih   = (const float*)d_in[5];
    const float* b_hh   = (const float*)d_in[6];
    const float* fc_w   = (const float*)d_in[7];
    const float* fc_b   = (const float*)d_in[8];

    float* gi = (float*)d_ws;          // 1536 floats
    float* gh = gi + G3H;              // 1536 floats

    gru_gates_wmma<<<192, 256, 0, stream>>>(token, hidden, emb,
                                            w_ih, w_hh, b_ih, b_hh, gi, gh);
    gru_tail<<<1, HID, 0, stream>>>(hidden, fc_w, fc_b, gi, gh, (float*)d_out);
}